// TtMixtralAttention_72765335929180
// MI455X (gfx1250) — compile-verified
//
#include <hip/hip_runtime.h>
#include <hip/hip_bf16.h>

// Problem constants (from reference)
#define B_    32
#define NH_   32
#define NKV_  8
#define D_    128
#define HID_  4096
#define WIN_  2048
#define G_    4
#define QKV_COLS ((NH_ + 2 * NKV_) * D_)   // 6144
#define KSPLIT 4

typedef __attribute__((ext_vector_type(2))) float v2f;
typedef __attribute__((ext_vector_type(8))) float v8f;

// One V_WMMA_F32_16X16X4_F32 step.
// A (16x4 f32): lane holds row M=lane%16; VGPR0 = K = (lane<16?0:2), VGPR1 = K+1.
// B (4x16 f32): lane holds col N=lane%16; VGPR0 = K = (lane<16?0:2), VGPR1 = K+1.
// C/D (16x16 f32): lane holds col N=lane%16; VGPR r holds row M = r + 8*(lane/16).
__device__ __forceinline__ v8f wmma_f32_k4(float a0, float a1, float b0, float b1, v8f c) {
  v2f A; A.x = a0; A.y = a1;
  v2f Bv; Bv.x = b0; Bv.y = b1;
  return __builtin_amdgcn_wmma_f32_16x16x4_f32(
      /*neg_a=*/false, A, /*neg_b=*/false, Bv,
      /*c_mod=*/(short)0, c, /*reuse_a=*/false, /*reuse_b=*/false);
}

// ---------------------------------------------------------------------------
// Kernel 1: partial xqkv = x(32x4096) @ wqkv(4096x6144), split-K by 4.
// Each wave: one 16-col tile, BOTH 16-row tiles (B fragments reused), 1/4 of K.
// 1536 wave-tiles -> 192 blocks. Deterministic partials, reduced later.
// ---------------------------------------------------------------------------
__global__ __launch_bounds__(256)
void qkv_gemm_kernel(const float* __restrict__ x, const float* __restrict__ wqkv,
                     float* __restrict__ part) {
  const int lane = threadIdx.x & 31;
  const int wave = threadIdx.x >> 5;
  const int tile = blockIdx.x * 8 + wave;      // 0..1535
  const int ct = tile & 383;                   // 6144 -> 384 col tiles
  const int ks = tile >> 9;                    // hmm: 1536/384 = 4 k-slices
  const int m16  = lane & 15;
  const int col  = ct * 16 + m16;
  const int koff = (lane < 16) ? 0 : 2;
  const int k0 = ks * (HID_ / KSPLIT);
  const int k1 = k0 + (HID_ / KSPLIT);
  const float* xr0 = x + (size_t)m16 * HID_;          // rows 0..15
  const float* xr1 = x + (size_t)(16 + m16) * HID_;   // rows 16..31
  v8f c0 = {}, c1 = {};
  for (int kk = k0; kk < k1; kk += 4) {
    const float2 a0 = *(const float2*)(xr0 + kk + koff);
    const float2 a1 = *(const float2*)(xr1 + kk + koff);
    const float b0 = wqkv[(size_t)(kk + koff) * QKV_COLS + col];
    const float b1 = wqkv[(size_t)(kk + koff + 1) * QKV_COLS + col];
    __builtin_prefetch(&wqkv[(size_t)(kk + 64 + koff) * QKV_COLS + col], 0, 0);
    c0 = wmma_f32_k4(a0.x, a0.y, b0, b1, c0);
    c1 = wmma_f32_k4(a1.x, a1.y, b0, b1, c1);
  }
  float* pb = part + (size_t)ks * (B_ * QKV_COLS);
  const int rbase = 8 * (lane >> 4);
#pragma unroll
  for (int r = 0; r < 8; ++r) {
    pb[(size_t)(rbase + r) * QKV_COLS + col]      = c0[r];
    pb[(size_t)(16 + rbase + r) * QKV_COLS + col] = c1[r];
  }
}

// ---------------------------------------------------------------------------
// Reduce 4 split-K partials: dst[i] = sum_{s<4} src[s*stride + i], float4-wide.
// ---------------------------------------------------------------------------
__global__ __launch_bounds__(256)
void reduce4_kernel(const float* __restrict__ src, float* __restrict__ dst,
                    int stride /* elements per slice */) {
  const int i = blockIdx.x * 256 + threadIdx.x;   // float4 index
  const float4 a = ((const float4*)src)[i];
  const float4 b = ((const float4*)(src + stride))[i];
  const float4 c = ((const float4*)(src + 2 * stride))[i];
  const float4 d = ((const float4*)(src + 3 * stride))[i];
  float4 o;
  o.x = a.x + b.x + c.x + d.x;
  o.y = a.y + b.y + c.y + d.y;
  o.z = a.z + b.z + c.z + d.z;
  o.w = a.w + b.w + c.w + d.w;
  ((float4*)dst)[i] = o;
}

// ---------------------------------------------------------------------------
// Kernel 2: rotary. Rows 0..1023 = q (b*NH+n), rows 1024..1279 = k (b*NKV+kv).
// qk_rot[row] = src_row(128) @ rot_mat(128x128).
// ---------------------------------------------------------------------------
__global__ __launch_bounds__(256)
void rope_gemm_kernel(const float* __restrict__ xqkv, const float* __restrict__ rot,
                      float* __restrict__ qk_rot) {
  const int lane = threadIdx.x & 31;
  const int wave = threadIdx.x >> 5;
  const int tile = blockIdx.x * 8 + wave;      // 80 blocks * 8 = 640 tiles
  const int ct = tile & 7;                     // 128 cols -> 8 tiles
  const int rt = tile >> 3;                    // 1280 rows -> 80 tiles
  const int m16  = lane & 15;
  const int col  = ct * 16 + m16;
  const int koff = (lane < 16) ? 0 : 2;

  const int row = rt * 16 + m16;
  const float* src;
  if (row < B_ * NH_) {                        // q rows
    const int b = row >> 5, n = row & 31;
    src = xqkv + (size_t)b * QKV_COLS + n * D_;
  } else {                                     // k rows
    const int r2 = row - B_ * NH_;
    const int b = r2 >> 3, kv = r2 & 7;
    src = xqkv + (size_t)b * QKV_COLS + NH_ * D_ + kv * D_;
  }
  v8f c = {};
  for (int kk = 0; kk < D_; kk += 4) {
    const float2 av = *(const float2*)(src + kk + koff);
    const float b0 = rot[(kk + koff) * D_ + col];
    const float b1 = rot[(kk + koff + 1) * D_ + col];
    c = wmma_f32_k4(av.x, av.y, b0, b1, c);
  }
  const int rbase = rt * 16 + 8 * (lane >> 4);
#pragma unroll
  for (int r = 0; r < 8; ++r)
    qk_rot[(size_t)(rbase + r) * D_ + col] = c[r];
}

// ---------------------------------------------------------------------------
// Kernel 3: attention for one (b, kv) per block (grid = 256).
//   q rows async-staged to LDS -> scores(4x2048) = qg @ K^T (WMMA) -> softmax
//   -> out = attn @ V (WMMA). New k/v rows (pos == current_pos) come from ws
//   instead of the caches (inputs must not be mutated).
// ---------------------------------------------------------------------------
__global__ __launch_bounds__(256)
void attn_kernel(const float* __restrict__ xqkv, const float* __restrict__ qk_rot,
                 const float* __restrict__ cache_k, const float* __restrict__ cache_v,
                 const int* __restrict__ p_start, const int* __restrict__ p_cur,
                 float* __restrict__ attn_out) {
  __shared__ float s[G_][WIN_];     // 32 KB of the 320 KB WGP LDS
  __shared__ float qs[G_][D_];      // 2 KB staged q rows
  __shared__ float red[256];

  const int tid  = threadIdx.x;
  const int lane = tid & 31;
  const int wave = tid >> 5;
  const int bk = blockIdx.x;                   // b*NKV + kv
  const int b = bk >> 3, kv = bk & 7;
  const int cur   = *p_cur;
  const int slice = min(*p_start + 1, WIN_);
  const int koff = (lane < 16) ? 0 : 2;
  const int m16  = lane & 15;
  const float scale = 0.08838834764831845f;    // 1/sqrt(128)

  const float* qbase  = qk_rot + (size_t)(b * NH_ + kv * G_) * D_;   // 4 q rows
  const float* knew   = qk_rot + (size_t)(B_ * NH_ + bk) * D_;
  const float* vnew   = xqkv + (size_t)b * QKV_COLS + (NH_ + NKV_) * D_ + kv * D_;
  const float* kcache = cache_k + (size_t)bk * WIN_ * D_;
  const float* vcache = cache_v + (size_t)bk * WIN_ * D_;

  // ---- Stage q rows (4x128 = 2KB) into LDS with gfx1250 async-to-LDS DMA.
  // Waves 0..3 issue one b128 per lane (wave-uniform guard keeps EXEC full).
  if (tid < 128) {
    const float* gsrc = qbase + tid * 4;
    const unsigned int ldst =
        (unsigned int)(uintptr_t)((const void*)(&qs[0][0] + tid * 4));
    asm volatile("global_load_async_to_lds_b128 %0, %1, off"
                 :: "v"(ldst), "v"(gsrc) : "memory");
  }
  asm volatile("s_wait_asynccnt 0x0" ::: "memory");
  __syncthreads();

  // ---- Phase 1: scores = q @ K^T (16 key positions per tile, K-dim = 128)
  for (int t = wave; t < WIN_ / 16; t += 8) {
    const int pos = t * 16 + m16;              // B-fragment column = key position
    const float* krow = (pos == cur) ? knew : (kcache + (size_t)pos * D_);
    v8f c = {};
    for (int kk = 0; kk < D_; kk += 4) {
      float a0 = 0.f, a1 = 0.f;
      if (m16 < G_) {                          // only 4 valid q rows, rest zero
        const float2 av = *(const float2*)(&qs[m16][kk + koff]);
        a0 = av.x; a1 = av.y;
      }
      const float2 bv = *(const float2*)(krow + kk + koff);
      c = wmma_f32_k4(a0, a1, bv.x, bv.y, c);
    }
    if (lane < 16) {                           // lanes 0..15 hold rows M=0..7
#pragma unroll
      for (int r = 0; r < G_; ++r)
        s[r][pos] = (pos < slice) ? c[r] * scale : -__builtin_inff();
    }
  }
  __syncthreads();

  // ---- Phase 2: row-wise softmax over 2048 (block-wide reductions)
  for (int r = 0; r < G_; ++r) {
    float lmax = -__builtin_inff();
    for (int j = tid; j < WIN_; j += 256) lmax = fmaxf(lmax, s[r][j]);
    red[tid] = lmax; __syncthreads();
    for (int off = 128; off > 0; off >>= 1) {
      if (tid < off) red[tid] = fmaxf(red[tid], red[tid + off]);
      __syncthreads();
    }
    const float rowmax = red[0]; __syncthreads();

    float lsum = 0.f;
    for (int j = tid; j < WIN_; j += 256) {
      const float e = __expf(s[r][j] - rowmax);
      s[r][j] = e; lsum += e;
    }
    red[tid] = lsum; __syncthreads();
    for (int off = 128; off > 0; off >>= 1) {
      if (tid < off) red[tid] += red[tid + off];
      __syncthreads();
    }
    const float inv = 1.0f / red[0]; __syncthreads();
    for (int j = tid; j < WIN_; j += 256) s[r][j] *= inv;
    __syncthreads();
  }

  // ---- Phase 3: out(4x128) = attn(4x2048) @ V(2048x128); 1 col-tile per wave
  const int col = wave * 16 + m16;             // 8 waves * 16 = 128 = D
  v8f o = {};
  for (int kk = 0; kk < WIN_; kk += 4) {
    const int p0 = kk + koff;
    float a0 = 0.f, a1 = 0.f;
    if (m16 < G_) {                            // LDS A-fragment (float2)
      const float2 sv = *(const float2*)(&s[m16][p0]);
      a0 = sv.x; a1 = sv.y;
    }
    const float* vr0 = (p0 == cur)     ? vnew : (vcache + (size_t)p0 * D_);
    const float* vr1 = (p0 + 1 == cur) ? vnew : (vcache + (size_t)(p0 + 1) * D_);
    o = wmma_f32_k4(a0, a1, vr0[col], vr1[col], o);
  }
  if (lane < 16) {
#pragma unroll
    for (int r = 0; r < G_; ++r)
      attn_out[(size_t)b * HID_ + (kv * G_ + r) * D_ + col] = o[r];
  }
}

// ---------------------------------------------------------------------------
// Kernel 4: partial dense = attn_out(32x4096) @ wo(4096x4096), split-K by 4,
// dual row tiles per wave. 1024 wave-tiles -> 128 blocks.
// ---------------------------------------------------------------------------
__global__ __launch_bounds__(256)
void dense_gemm_kernel(const float* __restrict__ ain, const float* __restrict__ wo,
                       float* __restrict__ part) {
  const int lane = threadIdx.x & 31;
  const int wave = threadIdx.x >> 5;
  const int tile = blockIdx.x * 8 + wave;      // 0..1023
  const int ct = tile & 255;                   // 4096 -> 256 col tiles
  const int ks = tile >> 8;                    // 4 k-slices
  const int m16  = lane & 15;
  const int col  = ct * 16 + m16;
  const int koff = (lane < 16) ? 0 : 2;
  const int k0 = ks * (HID_ / KSPLIT);
  const int k1 = k0 + (HID_ / KSPLIT);
  const float* ar0 = ain + (size_t)m16 * HID_;
  const float* ar1 = ain + (size_t)(16 + m16) * HID_;
  v8f c0 = {}, c1 = {};
  for (int kk = k0; kk < k1; kk += 4) {
    const float2 a0 = *(const float2*)(ar0 + kk + koff);
    const float2 a1 = *(const float2*)(ar1 + kk + koff);
    const float b0 = wo[(size_t)(kk + koff) * HID_ + col];
    const float b1 = wo[(size_t)(kk + koff + 1) * HID_ + col];
    __builtin_prefetch(&wo[(size_t)(kk + 64 + koff) * HID_ + col], 0, 0);
    c0 = wmma_f32_k4(a0.x, a0.y, b0, b1, c0);
    c1 = wmma_f32_k4(a1.x, a1.y, b0, b1, c1);
  }
  float* pb = part + (size_t)ks * (B_ * HID_);
  const int rbase = 8 * (lane >> 4);
#pragma unroll
  for (int r = 0; r < 8; ++r) {
    pb[(size_t)(rbase + r) * HID_ + col]      = c0[r];
    pb[(size_t)(16 + rbase + r) * HID_ + col] = c1[r];
  }
}

// ---------------------------------------------------------------------------
extern "C" void kernel_launch(void* const* d_in, const int* in_sizes, int n_in,
                              void* d_out, int out_size, void* d_ws, size_t ws_size,
                              hipStream_t stream) {
  const float* x       = (const float*)d_in[0];  // (1,1,32,4096)
  const float* wqkv    = (const float*)d_in[1];  // (4096,6144)
  const float* wo      = (const float*)d_in[2];  // (4096,4096)
  const float* rot     = (const float*)d_in[3];  // (128,128)
  const float* cache_k = (const float*)d_in[4];  // (32,8,2048,128)
  const float* cache_v = (const float*)d_in[5];  // (32,8,2048,128)
  const int*   p_start = (const int*)d_in[6];
  const int*   p_cur   = (const int*)d_in[7];

  float* ws     = (float*)d_ws;
  float* xqkv   = ws;                                            // 196608 f
  float* qk_rot = xqkv + (size_t)B_ * QKV_COLS;                  // 163840 f
  float* aout   = qk_rot + (size_t)(B_ * NH_ + B_ * NKV_) * D_;  // 131072 f
  float* part1  = aout + (size_t)B_ * HID_;                      // 4*196608 f
  float* part4  = part1 + (size_t)KSPLIT * B_ * QKV_COLS;        // 4*131072 f

  // 1) QKV projection (split-K partials + reduce)
  qkv_gemm_kernel<<<192, 256, 0, stream>>>(x, wqkv, part1);
  reduce4_kernel <<<192, 256, 0, stream>>>(part1, xqkv, B_ * QKV_COLS);
  // 2) rotary for q and k
  rope_gemm_kernel<<<80, 256, 0, stream>>>(xqkv, rot, qk_rot);
  // 3) GQA attention over the 2048-token window
  attn_kernel<<<256, 256, 0, stream>>>(xqkv, qk_rot, cache_k, cache_v,
                                       p_start, p_cur, aout);
  // 4) output projection (split-K partials + reduce)
  dense_gemm_kernel<<<128, 256, 0, stream>>>(aout, wo, part4);
  reduce4_kernel  <<<128, 256, 0, stream>>>(part4, (float*)d_out, B_ * HID_);
}